// ContrastiveLoss_58901181497927
// MI455X (gfx1250) — compile-verified
//
#include <hip/hip_runtime.h>
#include <hip/hip_bf16.h>

// ContrastiveLoss, MI455X / gfx1250.
// Algebraic collapse: (sim@mask)[i,j] = rowsum_i - classsum[i, labels_j], so
// neg = sum_i sum_c count[c]*exp(rowsum_i - classsum[i,c]) with
// classsum = txt_n @ imgclass^T, a [4096 x 128pad] x K512 f32 GEMM done with
// V_WMMA_F32_16X16X4_F32. Memory-bound: ~17MB mandatory traffic @ 23.3 TB/s.

typedef __attribute__((ext_vector_type(2))) float v2f;
typedef __attribute__((ext_vector_type(8))) float v8f;

#define NN 4096
#define DD 512
#define NC 100
#define NCP 128   // padded class count (8 tiles of 16)

// ---------------- Kernel 1: per-row norms + pos dot ----------------
__global__ void __launch_bounds__(256)
rowstats_kernel(const float* __restrict__ img, const float* __restrict__ txt,
                float* __restrict__ inv_img, float* __restrict__ inv_txt,
                float* __restrict__ posdot) {
    int i = blockIdx.x;
    int t = threadIdx.x;
    size_t base = (size_t)i * DD;
    float a0 = img[base + t],       b0 = txt[base + t];
    float a1 = img[base + 256 + t], b1 = txt[base + 256 + t];
    float si = a0 * a0 + a1 * a1;
    float st = b0 * b0 + b1 * b1;
    float sd = a0 * b0 + a1 * b1;
    for (int o = 16; o; o >>= 1) {
        si += __shfl_xor(si, o, 32);
        st += __shfl_xor(st, o, 32);
        sd += __shfl_xor(sd, o, 32);
    }
    __shared__ float r[3][8];
    int w = t >> 5;
    if ((t & 31) == 0) { r[0][w] = si; r[1][w] = st; r[2][w] = sd; }
    __syncthreads();
    if (t == 0) {
        float SI = 0.f, ST = 0.f, SD = 0.f;
        for (int j = 0; j < 8; j++) { SI += r[0][j]; ST += r[1][j]; SD += r[2][j]; }
        float ii = rsqrtf(SI), it = rsqrtf(ST);
        inv_img[i] = ii;
        inv_txt[i] = it;
        posdot[i]  = SD * ii * it;   // <img_n_i, txt_n_i>
    }
}

// ---------------- Kernel 2: per-class sums of normalized img rows ----------------
// Deterministic: ordered match list (sorted by k) via per-thread chunk scan +
// serial prefix; fixed accumulation order per output column.
__global__ void __launch_bounds__(256)
classsum_kernel(const float* __restrict__ img, const int* __restrict__ labels,
                const float* __restrict__ inv_img,
                float* __restrict__ imgclass, float* __restrict__ countf) {
    int c = blockIdx.x, t = threadIdx.x;
    if (c >= NC) {  // zero padding classes 100..127
        imgclass[(size_t)c * DD + t]       = 0.f;
        imgclass[(size_t)c * DD + 256 + t] = 0.f;
        if (t == 0) countf[c] = 0.f;
        return;
    }
    __shared__ int list[NN];
    __shared__ int cnt[256];
    __shared__ int off[257];
    int k0 = t * 16;
    int local[16];
    int m = 0;
    #pragma unroll
    for (int j = 0; j < 16; j++) {
        int k = k0 + j;
        if (labels[k] == c) local[m++] = k;
    }
    cnt[t] = m;
    __syncthreads();
    if (t == 0) {
        int run = 0;
        for (int j = 0; j < 256; j++) { off[j] = run; run += cnt[j]; }
        off[256] = run;
    }
    __syncthreads();
    int o = off[t];
    for (int j = 0; j < m; j++) list[o + j] = local[j];
    __syncthreads();
    int n = off[256];
    float a0 = 0.f, a1 = 0.f;
    for (int j = 0; j < n; j++) {
        int k = list[j];
        float s = inv_img[k];
        a0 += img[(size_t)k * DD + t]       * s;
        a1 += img[(size_t)k * DD + 256 + t] * s;
    }
    imgclass[(size_t)c * DD + t]       = a0;
    imgclass[(size_t)c * DD + 256 + t] = a1;
    if (t == 0) countf[c] = (float)n;
}

// ---------------- Kernel 3: WMMA GEMM  P[i,c] = txt_n_i . imgclass_c ----------------
// 8 waves/block, wave w -> class tile w (16 classes); block -> 16 txt rows.
// A (16x4 f32) layout: lane L holds row (L&15), K = 2*(L>>4)+{0,1}  -> one float2.
// B (4x16)   layout: lane L holds col (L&15), K = 2*(L>>4)+{0,1}  -> one float2
//   of imgclass row (class) since B = imgclass^T.
__global__ void __launch_bounds__(256)
gemm_pc_kernel(const float* __restrict__ txt, const float* __restrict__ inv_txt,
               const float* __restrict__ imgclass, float* __restrict__ P) {
    int wave = threadIdx.x >> 5;          // class tile 0..7
    int lane = threadIdx.x & 31;
    int row0 = blockIdx.x * 16;
    int m    = lane & 15;
    int koff = (lane >> 4) << 1;          // 0 or 2
    const float* arow = txt      + (size_t)(row0 + m)      * DD + koff;
    const float* brow = imgclass + (size_t)(wave * 16 + m) * DD + koff;
    v8f acc = {};
    #pragma unroll 8
    for (int k = 0; k < DD; k += 4) {
        v2f a = *(const v2f*)(arow + k);
        v2f b = *(const v2f*)(brow + k);
        acc = __builtin_amdgcn_wmma_f32_16x16x4_f32(
            /*neg_a=*/false, a, /*neg_b=*/false, b,
            /*c_mod=*/(short)0, acc, /*reuse_a=*/false, /*reuse_b=*/false);
    }
    // D layout: VGPR r: lanes 0-15 -> M=r, lanes 16-31 -> M=r+8; N = lane&15.
    int rbase = row0 + ((lane >> 4) << 3);
    int col   = wave * 16 + m;
    #pragma unroll
    for (int r = 0; r < 8; r++) {
        int row = rbase + r;
        P[(size_t)row * NCP + col] = acc[r] * inv_txt[row];  // fold txt row scale
    }
}

// ---------------- Kernel 4: negrow[i] = sum_c count[c]*exp(rowsum_i - P[i,c]) ----------------
__global__ void __launch_bounds__(128)
negrow_kernel(const float* __restrict__ P, const float* __restrict__ countf,
              float* __restrict__ negrow) {
    int i = blockIdx.x, t = threadIdx.x;
    __shared__ float red[4];
    float p  = P[(size_t)i * NCP + t];
    float v  = (t < NC) ? p : 0.f;        // rowsum over real classes only
    for (int o = 16; o; o >>= 1) v += __shfl_xor(v, o, 32);
    if ((t & 31) == 0) red[t >> 5] = v;
    __syncthreads();
    float rowsum = red[0] + red[1] + red[2] + red[3];
    float e = (t < NC) ? countf[t] * expf(rowsum - p) : 0.f;
    for (int o = 16; o; o >>= 1) e += __shfl_xor(e, o, 32);
    __syncthreads();
    if ((t & 31) == 0) red[t >> 5] = e;
    __syncthreads();
    if (t == 0) negrow[i] = red[0] + red[1] + red[2] + red[3];
}

// ---------------- Kernel 5: loss = sum_i log1p(neg * exp(-posdot_i)) ----------------
__global__ void __launch_bounds__(256)
final_kernel(const float* __restrict__ negrow, const float* __restrict__ posdot,
             float* __restrict__ out) {
    __shared__ float red[8];
    __shared__ float negsh;
    int t = threadIdx.x;
    float s = 0.f;
    for (int i = t; i < NN; i += 256) s += negrow[i];
    for (int o = 16; o; o >>= 1) s += __shfl_xor(s, o, 32);
    if ((t & 31) == 0) red[t >> 5] = s;
    __syncthreads();
    if (t == 0) {
        float n = 0.f;
        for (int j = 0; j < 8; j++) n += red[j];
        negsh = n;
    }
    __syncthreads();
    float neg = negsh;
    float l = 0.f;
    for (int i = t; i < NN; i += 256)
        l += log1pf(neg * expf(-posdot[i]));   // log((pos+neg)/pos)
    for (int o = 16; o; o >>= 1) l += __shfl_xor(l, o, 32);
    __syncthreads();
    if ((t & 31) == 0) red[t >> 5] = l;
    __syncthreads();
    if (t == 0) {
        float L = 0.f;
        for (int j = 0; j < 8; j++) L += red[j];
        out[0] = L;
    }
}

extern "C" void kernel_launch(void* const* d_in, const int* in_sizes, int n_in,
                              void* d_out, int out_size, void* d_ws, size_t ws_size,
                              hipStream_t stream) {
    const float* img    = (const float*)d_in[0];
    const float* txt    = (const float*)d_in[1];
    const int*   labels = (const int*)d_in[2];
    float*       out    = (float*)d_out;

    // Workspace layout (floats): ~2.45 MB total.
    float* ws       = (float*)d_ws;
    float* inv_img  = ws;                  // 4096
    float* inv_txt  = ws + 4096;           // 4096
    float* posdot   = ws + 8192;           // 4096
    float* countf   = ws + 12288;          // 128
    float* negrow   = ws + 12544;          // 4096
    float* imgclass = ws + 20480;          // 128*512 = 65536
    float* P        = ws + 86016;          // 4096*128 = 524288

    rowstats_kernel<<<NN, 256, 0, stream>>>(img, txt, inv_img, inv_txt, posdot);
    classsum_kernel<<<NCP, 256, 0, stream>>>(img, labels, inv_img, imgclass, countf);
    gemm_pc_kernel<<<NN / 16, 256, 0, stream>>>(txt, inv_txt, imgclass, P);
    negrow_kernel<<<NN, 128, 0, stream>>>(P, countf, negrow);
    final_kernel<<<1, 256, 0, stream>>>(negrow, posdot, out);
}